// GRU_20031727468931
// MI455X (gfx1250) — compile-verified
//
#include <hip/hip_runtime.h>
#include <hip/hip_bf16.h>
#include <cstddef>

// ---------------- problem constants ----------------
#define T_     16384
#define IN_    14
#define H_     100
#define G3_    300      // 3*H
#define LDG_   304      // padded row stride for gi buffers (19 tiles of 16)

typedef __attribute__((ext_vector_type(2))) float v2f;
typedef __attribute__((ext_vector_type(8))) float v8f;

// =====================================================================
// WMMA fp32 GEMM with bias:  C[m][n] = sum_k A[m][k] * W[n][k] + bias[n]
//   A: [M x K] row-major (lda), W: [N x K] row-major (ldw)  (i.e. x @ W^T)
//   C: [M x ldc], M must be a multiple of 16; N guarded (padded tiles).
// Uses V_WMMA_F32_16X16X4_F32; one 16x16 output tile per wave.
// All fragment loads are address-clamped + value-selected (no divergent
// branches feeding the WMMA chain; EXEC stays all-1s throughout).
// Fragment layouts per CDNA5 ISA 7.12.2:
//   A 16x4 f32 : lane = M (mod 16); half (lane>=16) selects K pair {0,1}/{2,3}
//   B 4x16 f32 : lane = N (mod 16); half selects K pair
//   C/D        : vgpr i -> M = i + 8*half, N = lane&15
// =====================================================================
template<int K>   // K must be even
__global__ __launch_bounds__(128)
void wmma_gemm_bias(const float* __restrict__ A, int lda,
                    const float* __restrict__ W, int ldw,
                    const float* __restrict__ bias,
                    float* __restrict__ C, int ldc,
                    int M, int N)
{
    const int lane = threadIdx.x & 31;
    const int wave = threadIdx.x >> 5;
    const int ntn  = (N + 15) >> 4;
    const int tile = blockIdx.x * (blockDim.x >> 5) + wave;
    if (tile >= (M >> 4) * ntn) return;          // wave-uniform: EXEC stays all-1

    const int m0    = (tile / ntn) << 4;
    const int n0    = (tile % ntn) << 4;
    const int half  = lane >> 4;                 // 0: K{0,1}, 1: K{2,3}
    const int l15   = lane & 15;
    const int arow  = m0 + l15;
    const int bcol  = n0 + l15;
    const bool bval = (bcol < N);
    const int bcolc = bval ? bcol : (N - 1);     // clamped: loads always legal

    // bias broadcast over M (unconditional load + select, no branch)
    const float bld = bias[bcolc];
    const float bv  = bval ? bld : 0.0f;
    v8f c;
#pragma unroll
    for (int i = 0; i < 8; ++i) c[i] = bv;

    const float* Arow = A + (size_t)arow  * lda;
    const float* Wrow = W + (size_t)bcolc * ldw;

#pragma unroll
    for (int k0 = 0; k0 < K; k0 += 4) {
        const int  ka  = k0 + half * 2;          // even
        const bool kok = (ka < K);               // K even => covers ka+1 too
        const int  kc  = kok ? ka : (K - 2);     // clamped, 8B-aligned

        const v2f av = *(const v2f*)(Arow + kc); // global_load_b64
        const v2f wv = *(const v2f*)(Wrow + kc); // global_load_b64
        const bool bok = kok && bval;

        v2f a, b;
        a.x = kok ? av.x : 0.0f;
        a.y = kok ? av.y : 0.0f;
        b.x = bok ? wv.x : 0.0f;
        b.y = bok ? wv.y : 0.0f;

        // D = A x B + C   (8 args: neg_a, A, neg_b, B, c_mod, C, reuse_a, reuse_b)
        c = __builtin_amdgcn_wmma_f32_16x16x4_f32(false, a, false, b,
                                                  (short)0, c, false, false);
    }

    if (bval) {
#pragma unroll
        for (int i = 0; i < 8; ++i)
            C[(size_t)(m0 + i + half * 8) * ldc + bcol] = c[i];
    }
}

// =====================================================================
// Persistent single-workgroup GRU recurrence (one layer).
//   gi   : [T x LDG_] precomputed input projections incl. b_ih (gate order r,z,n)
//   w_hh : [300 x 100], b_hh : [300]
//   h_seq: [T x 100] output hidden states (may be null)
//   fc_* : if fc_w != null, emit final scalar out = h_T . fc_w + fc_b
// Recurrent weights live in LDS (120 KB of the WGP's 320 KB); the h vector
// is broadcast from LDS each step. 384 threads: 300 do the matvec rows,
// 100 do the gate math.
// =====================================================================
__global__ __launch_bounds__(384)
void gru_rec_kernel(const float* __restrict__ gi,
                    const float* __restrict__ w_hh,
                    const float* __restrict__ b_hh,
                    float* __restrict__ h_seq,
                    const float* __restrict__ fc_w,
                    const float* __restrict__ fc_b,
                    float* __restrict__ out)
{
    extern __shared__ float smem[];
    float* s_w  = smem;            // 30000 floats (w_hh), 16B-aligned rows (400B)
    float* s_h  = smem + 30000;    // 104 floats (h, padded)
    float* s_gh = smem + 30104;    // 300 floats (gate pre-activations)
    float* s_b  = smem + 30404;    // 300 floats (b_hh)

    const int tid = threadIdx.x;

    for (int i = tid; i < G3_ * H_; i += 384) s_w[i] = w_hh[i];
    if (tid < G3_)  s_b[tid] = b_hh[tid];
    if (tid < 104)  s_h[tid] = 0.0f;
    __syncthreads();

    for (int t = 0; t < T_; ++t) {
        // prefetch gi row t+8 (1216 B -> 10 cachelines)
        if (tid < 10) {
            const int tp = t + 8;
            if (tp < T_)
                __builtin_prefetch(gi + (size_t)tp * LDG_ + tid * 32, 0, 1);
        }

        // gh = w_hh @ h + b_hh   (300 rows, dot length 100, float4 LDS loads)
        if (tid < G3_) {
            const float4* wr = (const float4*)(s_w + tid * H_);
            const float4* hv = (const float4*)s_h;
            float acc = 0.0f;
#pragma unroll
            for (int k = 0; k < H_ / 4; ++k) {
                const float4 w4 = wr[k];
                const float4 h4 = hv[k];
                acc = fmaf(w4.x, h4.x, acc);
                acc = fmaf(w4.y, h4.y, acc);
                acc = fmaf(w4.z, h4.z, acc);
                acc = fmaf(w4.w, h4.w, acc);
            }
            s_gh[tid] = acc + s_b[tid];
        }
        __syncthreads();

        // gates + state update
        if (tid < H_) {
            const float* girow = gi + (size_t)t * LDG_;
            const float i_r = girow[tid];
            const float i_z = girow[H_ + tid];
            const float i_n = girow[2 * H_ + tid];
            const float h_r = s_gh[tid];
            const float h_z = s_gh[H_ + tid];
            const float h_n = s_gh[2 * H_ + tid];
            const float r = 1.0f / (1.0f + __expf(-(i_r + h_r)));
            const float z = 1.0f / (1.0f + __expf(-(i_z + h_z)));
            const float n = tanhf(i_n + r * h_n);
            const float hnew = (1.0f - z) * n + z * s_h[tid];
            s_h[tid] = hnew;
            if (h_seq) h_seq[(size_t)t * H_ + tid] = hnew;
        }
        __syncthreads();
    }

    // final FC on the last hidden state (layer-1 invocation only)
    if (fc_w) {
        if (tid < H_) s_gh[tid] = s_h[tid] * fc_w[tid];
        __syncthreads();
        if (tid == 0) {
            float s = fc_b[0];
            for (int j = 0; j < H_; ++j) s += s_gh[j];
            out[0] = s;
        }
    }
}

// =====================================================================
extern "C" void kernel_launch(void* const* d_in, const int* in_sizes, int n_in,
                              void* d_out, int out_size, void* d_ws, size_t ws_size,
                              hipStream_t stream)
{
    const float* x     = (const float*)d_in[0];
    const float* w_ih0 = (const float*)d_in[1];
    const float* w_hh0 = (const float*)d_in[2];
    const float* b_ih0 = (const float*)d_in[3];
    const float* b_hh0 = (const float*)d_in[4];
    const float* w_ih1 = (const float*)d_in[5];
    const float* w_hh1 = (const float*)d_in[6];
    const float* b_ih1 = (const float*)d_in[7];
    const float* b_hh1 = (const float*)d_in[8];
    const float* fc_w  = (const float*)d_in[9];
    const float* fc_b  = (const float*)d_in[10];
    float* out = (float*)d_out;

    // workspace carve: gi buffer is reused for layer-0 and layer-1 projections
    float* gi = (float*)d_ws;                      // [T x LDG_]
    float* h1 = gi + (size_t)T_ * LDG_;            // [T x H_]

    const int mtiles = T_ / 16;                    // 1024
    const int ntiles = (G3_ + 15) / 16;            // 19
    const int tiles  = mtiles * ntiles;            // 19456 wave-tiles
    const dim3 blk(128);                           // 4 waves / block
    const dim3 grd((tiles + 3) / 4);

    const size_t smem_bytes = (size_t)(30404 + 300) * sizeof(float); // ~123 KB
    (void)hipFuncSetAttribute((const void*)gru_rec_kernel,
                              hipFuncAttributeMaxDynamicSharedMemorySize,
                              (int)smem_bytes);

    // layer 0: input projections (K=14 padded to 16 inside the kernel)
    wmma_gemm_bias<IN_><<<grd, blk, 0, stream>>>(x, IN_, w_ih0, IN_, b_ih0,
                                                 gi, LDG_, T_, G3_);
    // layer 0: recurrence -> h1
    gru_rec_kernel<<<1, 384, smem_bytes, stream>>>(gi, w_hh0, b_hh0, h1,
                                                   nullptr, nullptr, nullptr);
    // layer 1: input projections gi1 = h1 @ w_ih1^T + b_ih1 (K=100, 25 WMMAs)
    wmma_gemm_bias<H_><<<grd, blk, 0, stream>>>(h1, H_, w_ih1, H_, b_ih1,
                                                gi, LDG_, T_, G3_);
    // layer 1: recurrence + final FC -> out[0]
    gru_rec_kernel<<<1, 384, smem_bytes, stream>>>(gi, w_hh1, b_hh1, nullptr,
                                                   fc_w, fc_b, out);
}